// CGMM_39015482917006
// MI455X (gfx1250) — compile-verified
//
#include <hip/hip_runtime.h>
#include <hip/hip_bf16.h>
#include <math.h>

// Problem constants (match reference file)
#define C_HID   16
#define M_LAB   32
#define N_GEN   16

typedef __attribute__((ext_vector_type(2))) float v2f;
typedef __attribute__((ext_vector_type(8))) float v8f;

// ---------------------------------------------------------------------------
// Kernel 1: zero the histogram scratch (ws is poisoned / left dirty by us)
// ---------------------------------------------------------------------------
__global__ void cgmm_zero(float* __restrict__ p, int n) {
    int i = blockIdx.x * blockDim.x + threadIdx.x;
    if (i < n) p[i] = 0.0f;
}

// ---------------------------------------------------------------------------
// Kernel 2: build the 32x16 negated-likelihood lookup table.
// One block, 512 threads: thread (m,g) owns table[m][g].
//   lseB[c][g]  = logsumexp_m B[c,m,g]   (softmax over labels)
//   lsePi[g]    = logsumexp_c Pi[c,g]    (softmax over hidden states)
//   ln_c        = (B[c,m,g]-lseB) + (Pi[c,g]-lsePi) = log(numerator_c)
//   table[m][g] = -(sum_c e^ln * ln) / (sum_c e^ln)
// ---------------------------------------------------------------------------
__global__ __launch_bounds__(512) void cgmm_table(const float* __restrict__ B,
                                                  const float* __restrict__ Pi,
                                                  float* __restrict__ table) {
    __shared__ float lseB[C_HID][N_GEN];
    __shared__ float lsePi[N_GEN];
    const int tid = threadIdx.x;

    if (tid < C_HID * N_GEN) {              // 256 threads: (c,g) pairs
        const int c = tid >> 4, g = tid & 15;
        float mx = -1e30f;
        for (int m = 0; m < M_LAB; ++m)
            mx = fmaxf(mx, B[c * (M_LAB * N_GEN) + m * N_GEN + g]);
        float s = 0.0f;
        for (int m = 0; m < M_LAB; ++m)
            s += expf(B[c * (M_LAB * N_GEN) + m * N_GEN + g] - mx);
        lseB[c][g] = mx + logf(s);
    } else if (tid < C_HID * N_GEN + N_GEN) {  // 16 threads: per-gen Pi lse
        const int g = tid - C_HID * N_GEN;
        float mx = -1e30f;
        for (int c = 0; c < C_HID; ++c) mx = fmaxf(mx, Pi[c * N_GEN + g]);
        float s = 0.0f;
        for (int c = 0; c < C_HID; ++c) s += expf(Pi[c * N_GEN + g] - mx);
        lsePi[g] = mx + logf(s);
    }
    __syncthreads();

    const int m = tid >> 4;                 // label 0..31
    const int g = tid & 15;                 // generator 0..15
    float s = 0.0f, acc = 0.0f;
    for (int c = 0; c < C_HID; ++c) {
        float ln = (B[c * (M_LAB * N_GEN) + m * N_GEN + g] - lseB[c][g]) +
                   (Pi[c * N_GEN + g] - lsePi[g]);
        float nu = expf(ln);
        s   += nu;
        acc += nu * ln;
    }
    table[m * N_GEN + g] = -(acc / s);      // pre-negated: out = cnt @ table
}

// ---------------------------------------------------------------------------
// Kernel 3: per-graph label histogram.  batch is sorted but atomics on float
// counters in L2 are plenty for 500K increments spread over 160K counters.
// ---------------------------------------------------------------------------
__global__ void cgmm_hist(const int* __restrict__ x,
                          const int* __restrict__ batch,
                          float* __restrict__ hist, int n) {
    int i = blockIdx.x * blockDim.x + threadIdx.x;
    if (i < n) {
        atomicAdd(&hist[batch[i] * M_LAB + x[i]], 1.0f);
    }
}

// ---------------------------------------------------------------------------
// Kernel 4: out(5000x16) = hist(5000x32) @ table(32x16) via
// V_WMMA_F32_16X16X4_F32.  One wave per 16-graph tile, K=32 in 8 steps.
//   A (16x4 f32): lane = (m = lane&15, khalf = lane>>4);
//                 VGPR0 = A[m][kb+2*khalf], VGPR1 = A[m][kb+2*khalf+1]
//   B (4x16 f32): mirrored — VGPR0 = B[kb+2*khalf][n], VGPR1 = next K row
//   C/D (16x16) : VGPR r -> row (r + 8*khalf), col = lane&15
// All 32 lanes active (EXEC all-1s); guards only clamp/skip at the edges.
// ---------------------------------------------------------------------------
__global__ __launch_bounds__(32) void cgmm_gemm(const float* __restrict__ hist,
                                                const float* __restrict__ table,
                                                float* __restrict__ out,
                                                int n_graphs) {
    const int lane  = threadIdx.x;       // 0..31
    const int col   = lane & 15;         // A row (graph-in-tile) and B/D col
    const int khalf = lane >> 4;         // 0 or 1
    const int gbase = blockIdx.x * 16;

    int arow = gbase + col;
    if (arow > n_graphs - 1) arow = n_graphs - 1;   // clamp: padded rows unused

    v8f acc = {};
#pragma unroll
    for (int kb = 0; kb < M_LAB; kb += 4) {
        const int k0 = kb + 2 * khalf;
        // A: two consecutive K elements of this graph's histogram row (8B load)
        v2f a = *(const v2f*)(hist + arow * M_LAB + k0);
        // B: table rows k0 and k0+1, column `col`
        v2f b;
        b.x = table[k0 * N_GEN + col];
        b.y = table[(k0 + 1) * N_GEN + col];
        acc = __builtin_amdgcn_wmma_f32_16x16x4_f32(
            /*neg_a=*/false, a, /*neg_b=*/false, b,
            /*c_mod=*/(short)0, acc, /*reuse_a=*/false, /*reuse_b=*/false);
    }

#pragma unroll
    for (int r = 0; r < 8; ++r) {
        const int graph = gbase + r + 8 * khalf;
        if (graph < n_graphs) out[graph * N_GEN + col] = acc[r];
    }
}

// ---------------------------------------------------------------------------
// Launch: inputs are (x, edge_index, batch, B, Pi); edge_index is unused by
// layer 0 of the reference.  Output: 5000x1x16 floats.
// ws layout: [0, 5000*32) floats = histogram; next 32*16 floats = table.
// ---------------------------------------------------------------------------
extern "C" void kernel_launch(void* const* d_in, const int* in_sizes, int n_in,
                              void* d_out, int out_size, void* d_ws, size_t ws_size,
                              hipStream_t stream) {
    const int*   x     = (const int*)  d_in[0];
    const int*   batch = (const int*)  d_in[2];
    const float* B     = (const float*)d_in[3];
    const float* Pi    = (const float*)d_in[4];
    float*       out   = (float*)d_out;

    const int n_nodes  = in_sizes[0];
    const int n_graphs = out_size / N_GEN;          // 5000

    float* hist  = (float*)d_ws;                    // n_graphs * 32 floats
    float* table = hist + (size_t)n_graphs * M_LAB; // 32 * 16 floats

    const int hist_elems = n_graphs * M_LAB;
    cgmm_zero<<<(hist_elems + 255) / 256, 256, 0, stream>>>(hist, hist_elems);
    cgmm_table<<<1, 512, 0, stream>>>(B, Pi, table);
    cgmm_hist<<<(n_nodes + 255) / 256, 256, 0, stream>>>(x, batch, hist, n_nodes);
    cgmm_gemm<<<(n_graphs + 15) / 16, 32, 0, stream>>>(hist, table, out, n_graphs);
}